// FeedbackAttention_33535104648038
// MI455X (gfx1250) — compile-verified
//
#include <hip/hip_runtime.h>
#include <math.h>

#define SEQ   2048
#define BATCH 32
#define H     16
#define DK    64
#define DM    1024
#define PP    4096

typedef float v2f __attribute__((ext_vector_type(2)));
typedef float v8f __attribute__((ext_vector_type(8)));
typedef int   v4i __attribute__((ext_vector_type(4)));

// ---- CDNA5 async global->LDS copy (ASYNCcnt path), with sync fallback ----
#ifndef USE_ASYNC
#if defined(__has_builtin)
#if __has_builtin(__builtin_amdgcn_global_load_async_to_lds_b128) && \
    __has_builtin(__builtin_amdgcn_s_wait_asynccnt)
#define USE_ASYNC 1
#endif
#endif
#endif
#ifndef USE_ASYNC
#define USE_ASYNC 0
#endif

#if USE_ASYNC
__device__ __forceinline__ void cp16_async(void* lds, const void* g) {
  __builtin_amdgcn_global_load_async_to_lds_b128(
      (__attribute__((address_space(1))) v4i*)g,
      (__attribute__((address_space(3))) v4i*)lds,
      0, 0);
}
__device__ __forceinline__ void lds_ready() {
  __builtin_amdgcn_s_wait_asynccnt(0);   // our issued copies have landed in LDS
  __syncthreads();                       // publish across waves
}
#else
__device__ __forceinline__ void lds_ready() { __syncthreads(); }
#endif

// ---------------- workspace layout (float offsets) ----------------
#define OFF_Q   0
#define OFF_QT  (32 * 1024)
#define OFF_U   (2 * 32 * 1024)
#define OFF_SC  (OFF_U + 32 * 16 * 1024)       // scores (ac) then attn, in place
#define OFF_W   (OFF_SC + SEQ * BATCH * H)
#define OFF_X   (OFF_W + 32 * 16 * 1024)

// 1) q = query @ Wq ; qt = q + query_pos_bias (broadcast over b)
__global__ void __launch_bounds__(256) k_qproj(
    const float* __restrict__ query, const float* __restrict__ Wq,
    const float* __restrict__ qpb, float* __restrict__ q, float* __restrict__ qt) {
  int idx = blockIdx.x * blockDim.x + threadIdx.x;  // 32*1024
  int b = idx >> 10, n = idx & 1023;
  const float* qr = query + b * DM;
  float acc = 0.f;
  for (int c = 0; c < DM; ++c) acc = fmaf(qr[c], Wq[c * DM + n], acc);
  q[idx]  = acc;
  qt[idx] = acc + qpb[n];   // qpb flat[(h,d)] == flat[n]
}

// 2) u[b,h,c] = sum_d Wk[c, h*64+d] * qt[b,h,d]
__global__ void __launch_bounds__(256) k_u(
    const float* __restrict__ Wk, const float* __restrict__ qt, float* __restrict__ u) {
  int idx = blockIdx.x * blockDim.x + threadIdx.x;  // idx = (b*16+h)*1024+c
  int c = idx & 1023;
  int h = (idx >> 10) & 15;
  int b = idx >> 14;
  const float* wr = Wk + (size_t)c * DM + h * DK;
  const float* qr = qt + b * DM + h * DK;
  float acc = 0.f;
#pragma unroll
  for (int d = 0; d < DK; ++d) acc = fmaf(wr[d], qr[d], acc);
  u[idx] = acc;
}

// 3) ac[j,b,h] = sum_c key[j,b,c] * u[b,h,c]   (WMMA f32 16x16x4, double-buffered)
//    LDS row stride 36 words: 144B (16B aligned for b128) and 36*m mod 64 is a
//    conflict-free permutation over the 16 fragment lanes.
__global__ void __launch_bounds__(256) k_scores_wmma(
    const float* __restrict__ key, const float* __restrict__ u, float* __restrict__ scores) {
  const int j0 = blockIdx.x * 128;   // 16 j-tiles
  const int b  = blockIdx.y;         // 32
  __shared__ float klds[2][128 * 36];
  __shared__ float ulds[2][16 * 36];
  const int tid = threadIdx.x;
  const int wv = tid >> 5, lane = tid & 31;
  const float* keyb = key + ((size_t)j0 * BATCH + b) * DM;
  const float* ubase = u + (size_t)b * H * DM;

  auto load_chunk = [&](int buf, int c0) {
#if USE_ASYNC
#pragma unroll
    for (int i = 0; i < 4; ++i) {
      int v = tid + 256 * i;                 // 1024 vec4s: 128 rows x 8
      int row = v >> 3, q4 = (v & 7) << 2;
      cp16_async(&klds[buf][row * 36 + q4],
                 keyb + (size_t)row * (BATCH * DM) + c0 + q4);
    }
    if (tid < 128) {                         // 128 vec4s: 16 rows x 8
      int row = tid >> 3, q4 = (tid & 7) << 2;
      cp16_async(&ulds[buf][row * 36 + q4], ubase + row * DM + c0 + q4);
    }
#else
    for (int f = tid; f < 128 * 32; f += 256) {
      int row = f >> 5, cl = f & 31;
      klds[buf][row * 36 + cl] = keyb[(size_t)row * (BATCH * DM) + c0 + cl];
    }
    for (int f = tid; f < 16 * 32; f += 256) {
      int hh = f >> 5, cl = f & 31;
      ulds[buf][hh * 36 + cl] = ubase[hh * DM + c0 + cl];
    }
#endif
  };

  v8f acc = {};
  load_chunk(0, 0);
  // A 16x4 fp32 fragment: lane m=L%16 holds K = 2*(L/16)+{0,1}; B mirrors, n=L%16.
  const int m  = lane & 15;
  const int kb = (lane >> 4) << 1;
  for (int it = 0; it < DM / 32; ++it) {
    lds_ready();
    if (it + 1 < DM / 32) load_chunk((it + 1) & 1, (it + 1) * 32);
    const int cur = it & 1;
    const float* ka = &klds[cur][(wv * 16 + m) * 36 + kb];
    const float* uu = &ulds[cur][m * 36 + kb];
#pragma unroll
    for (int kk = 0; kk < 8; ++kk) {
      v2f a, bb;
      a.x  = ka[kk * 4];     a.y  = ka[kk * 4 + 1];
      bb.x = uu[kk * 4];     bb.y = uu[kk * 4 + 1];
      acc = __builtin_amdgcn_wmma_f32_16x16x4_f32(false, a, false, bb,
                                                  (short)0, acc, false, false);
    }
    __syncthreads();   // all waves done with buf[cur] before it is refilled
  }
  // D layout: vgpr i -> M = i + 8*(lane/16), N = lane%16
  const int n  = lane & 15;            // head
  const int mg = (lane >> 4) * 8;
#pragma unroll
  for (int i = 0; i < 8; ++i) {
    int j = j0 + wv * 16 + mg + i;
    scores[((size_t)j * BATCH + b) * H + n] = acc[i];
  }
}

// 4) scores = (ac + bd + kpb)*scale ; softmax over j ; write attn in place
__global__ void __launch_bounds__(256) k_softmax(
    const float* __restrict__ q, const float* __restrict__ kpe,
    const float* __restrict__ kpb, float* __restrict__ scores) {
  const int b = blockIdx.x >> 4;
  const int h = blockIdx.x & 15;
  __shared__ float qv[DK];
  __shared__ float red[256];
  const int tid = threadIdx.x;
  if (tid < DK) qv[tid] = q[b * DM + h * DK + tid];
  __syncthreads();
  const float scale = 0.125f;  // 1/sqrt(64)
  float s[8];
#pragma unroll
  for (int r = 0; r < 8; ++r) {
    int j  = tid + 256 * r;
    int pj = PP - SEQ + j;
    const float* kr = kpe + ((size_t)pj * H + h) * DK;
    float dot = 0.f;
#pragma unroll
    for (int d = 0; d < DK; ++d) dot = fmaf(qv[d], kr[d], dot);
    float ac = scores[((size_t)j * BATCH + b) * H + h];
    s[r] = (ac + dot + kpb[pj * H + h]) * scale;
  }
  float mx = s[0];
#pragma unroll
  for (int r = 1; r < 8; ++r) mx = fmaxf(mx, s[r]);
  red[tid] = mx; __syncthreads();
  for (int off = 128; off > 0; off >>= 1) {
    if (tid < off) red[tid] = fmaxf(red[tid], red[tid + off]);
    __syncthreads();
  }
  mx = red[0]; __syncthreads();
  float sum = 0.f;
#pragma unroll
  for (int r = 0; r < 8; ++r) { s[r] = __expf(s[r] - mx); sum += s[r]; }
  red[tid] = sum; __syncthreads();
  for (int off = 128; off > 0; off >>= 1) {
    if (tid < off) red[tid] += red[tid + off];
    __syncthreads();
  }
  float inv = 1.f / red[0];
#pragma unroll
  for (int r = 0; r < 8; ++r) {
    int j = tid + 256 * r;
    scores[((size_t)j * BATCH + b) * H + h] = s[r] * inv;
  }
}

// 5) w[b,h,c] = sum_j attn[j,b,h] * value[j,b,c]   (WMMA f32 16x16x4, double-buffered)
//    vlds stride 136 words: 544B (16B aligned); fragment lanes 16..31 offset by
//    +272 words = +16 banks -> disjoint from lanes 0..15 (conflict-free).
__global__ void __launch_bounds__(256) k_wsum_wmma(
    const float* __restrict__ value, const float* __restrict__ attn, float* __restrict__ w) {
  const int c0 = blockIdx.x * 128;   // 8 c-tiles
  const int b  = blockIdx.y;         // 32
  __shared__ float alds[2][32 * 16];    // attn chunk [j_local][h]
  __shared__ float vlds[2][32 * 136];   // value tile [j_local][c_local]
  const int tid = threadIdx.x;
  const int wv = tid >> 5, lane = tid & 31;

  auto load_chunk = [&](int buf, int j0) {
#if USE_ASYNC
#pragma unroll
    for (int i = 0; i < 4; ++i) {
      int v = tid + 256 * i;                 // 1024 vec4s: 32 rows x 32
      int row = v >> 5, q4 = (v & 31) << 2;
      cp16_async(&vlds[buf][row * 136 + q4],
                 value + ((size_t)(j0 + row) * BATCH + b) * DM + c0 + q4);
    }
    if (tid < 128) {                         // 128 vec4s: 32 rows x 4
      int row = tid >> 2, q4 = (tid & 3) << 2;
      cp16_async(&alds[buf][row * 16 + q4],
                 attn + ((size_t)(j0 + row) * BATCH + b) * H + q4);
    }
#else
    for (int f = tid; f < 32 * 128; f += 256) {
      int row = f >> 7, cl = f & 127;
      vlds[buf][row * 136 + cl] = value[((size_t)(j0 + row) * BATCH + b) * DM + c0 + cl];
    }
    for (int f = tid; f < 32 * 16; f += 256) {
      int row = f >> 4, hh = f & 15;
      alds[buf][row * 16 + hh] = attn[((size_t)(j0 + row) * BATCH + b) * H + hh];
    }
#endif
  };

  v8f acc = {};
  load_chunk(0, 0);
  const int m  = lane & 15;           // head (A rows) / c offset (B cols)
  const int kb = (lane >> 4) << 1;
  const int cB = wv * 16 + m;
  for (int it = 0; it < SEQ / 32; ++it) {
    lds_ready();
    if (it + 1 < SEQ / 32) load_chunk((it + 1) & 1, (it + 1) * 32);
    const int cur = it & 1;
#pragma unroll
    for (int kk = 0; kk < 8; ++kk) {
      int jb = kk * 4 + kb;
      v2f a, bb;
      a.x  = alds[cur][jb * 16 + m];
      a.y  = alds[cur][(jb + 1) * 16 + m];
      bb.x = vlds[cur][jb * 136 + cB];
      bb.y = vlds[cur][(jb + 1) * 136 + cB];
      acc = __builtin_amdgcn_wmma_f32_16x16x4_f32(false, a, false, bb,
                                                  (short)0, acc, false, false);
    }
    __syncthreads();   // all waves done with buf[cur] before it is refilled
  }
  const int n  = lane & 15;
  const int mg = (lane >> 4) * 8;
#pragma unroll
  for (int i = 0; i < 8; ++i) {
    int h = mg + i;
    w[((size_t)(b * H) + h) * DM + c0 + wv * 16 + n] = acc[i];
  }
}

// 6) x[b, h*64+d] = sum_c w[b,h,c] * Wv[c, h*64+d]
__global__ void __launch_bounds__(256) k_x(
    const float* __restrict__ w, const float* __restrict__ Wv, float* __restrict__ x) {
  int idx = blockIdx.x * blockDim.x + threadIdx.x;  // 32*1024
  int b = idx >> 10, n = idx & 1023;
  int h = n >> 6;
  const float* wr = w + ((size_t)(b * H) + h) * DM;
  float acc = 0.f;
  for (int c = 0; c < DM; ++c) acc = fmaf(wr[c], Wv[(size_t)c * DM + n], acc);
  x[idx] = acc;
}

// 7) out = x @ Wo + bo
__global__ void __launch_bounds__(256) k_out(
    const float* __restrict__ x, const float* __restrict__ Wo,
    const float* __restrict__ bo, float* __restrict__ out) {
  int idx = blockIdx.x * blockDim.x + threadIdx.x;  // 32*1024
  int b = idx >> 10, n = idx & 1023;
  const float* xr = x + b * DM;
  float acc = bo[n];
  for (int t = 0; t < DM; ++t) acc = fmaf(xr[t], Wo[(size_t)t * DM + n], acc);
  out[idx] = acc;
}

extern "C" void kernel_launch(void* const* d_in, const int* in_sizes, int n_in,
                              void* d_out, int out_size, void* d_ws, size_t ws_size,
                              hipStream_t stream) {
  const float* query = (const float*)d_in[0];
  const float* key   = (const float*)d_in[1];
  const float* value = (const float*)d_in[2];
  const float* Wq    = (const float*)d_in[3];
  const float* Wk    = (const float*)d_in[4];
  const float* Wv    = (const float*)d_in[5];
  const float* Wo    = (const float*)d_in[6];
  const float* bo    = (const float*)d_in[7];
  const float* kpe   = (const float*)d_in[8];
  const float* kpb   = (const float*)d_in[9];
  const float* qpb   = (const float*)d_in[10];
  float* out = (float*)d_out;

  float* ws = (float*)d_ws;
  float* q_buf  = ws + OFF_Q;
  float* qt_buf = ws + OFF_QT;
  float* u_buf  = ws + OFF_U;
  float* sc_buf = ws + OFF_SC;
  float* w_buf  = ws + OFF_W;
  float* x_buf  = ws + OFF_X;

  k_qproj<<<128, 256, 0, stream>>>(query, Wq, qpb, q_buf, qt_buf);
  k_u<<<2048, 256, 0, stream>>>(Wk, qt_buf, u_buf);
  k_scores_wmma<<<dim3(SEQ / 128, BATCH), 256, 0, stream>>>(key, u_buf, sc_buf);
  k_softmax<<<BATCH * H, 256, 0, stream>>>(q_buf, kpe, kpb, sc_buf);
  k_wsum_wmma<<<dim3(DM / 128, BATCH), 256, 0, stream>>>(value, sc_buf, w_buf);
  k_x<<<128, 256, 0, stream>>>(w_buf, Wv, x_buf);
  k_out<<<128, 256, 0, stream>>>(x_buf, Wo, bo, out);
}